// AutoRegressiveRNN_30571577213763
// MI455X (gfx1250) — compile-verified
//
#include <hip/hip_runtime.h>

typedef float v2f __attribute__((ext_vector_type(2)));
typedef float v8f __attribute__((ext_vector_type(8)));

#define B_TOT 1024
#define T_SEQ 13
#define Y_DIM 188
#define H_DIM 512
#define G_DIM 1536
#define NSTEP 13
#define BM 16
#define THREADS 512   // 16 waves; each wave owns 2 column tiles per gate

// LDS layout, in float2 units:
//   sx : 13 slots x 96 y-pairs x 16 batch rows   (circular x window, Y padded 188->192)
//   h0 : 256 h-pairs x 16 batch rows
//   h1 : 256 h-pairs x 16 batch rows
#define SX_F2 (T_SEQ * 96 * 16)   // 19968
#define H_F2  (256 * 16)          // 4096
#define LDS_F2 (SX_F2 + 2 * H_F2) // 28160 float2 = 225,280 bytes (< 320KB WGP LDS)

__device__ __forceinline__ v8f wmma4(v2f a, v2f b, v8f c) {
  // V_WMMA_F32_16X16X4_F32 : D(16x16 f32) = A(16x4 f32) * B(4x16 f32) + C
  return __builtin_amdgcn_wmma_f32_16x16x4_f32(false, a, false, b, (short)0, c,
                                               false, false);
}

__device__ __forceinline__ float sigm(float x) {
  return 1.0f / (1.0f + __expf(-x));
}
__device__ __forceinline__ float tanh_fast(float x) {
  float ax = fabsf(x);
  float e = __expf(-2.0f * ax);
  float r = (1.0f - e) / (1.0f + e);
  return copysignf(r, x);
}

// Load the 6 B operands (2 col tiles x 3 gates), all immediate offsets off one base.
__device__ __forceinline__ void ldb6(const v2f* __restrict__ b_p, v2f* b) {
  b[0] = b_p[0];
  b[1] = b_p[16];
  b[2] = b_p[H_DIM];
  b[3] = b_p[H_DIM + 16];
  b[4] = b_p[2 * H_DIM];
  b[5] = b_p[2 * H_DIM + 16];
}
// The 6 WMMA accumulations for one K-step.
__device__ __forceinline__ void mac6(v2f a, const v2f* b, v8f* aR, v8f* aZ,
                                     v8f* aN) {
  aR[0] = wmma4(a, b[0], aR[0]);
  aR[1] = wmma4(a, b[1], aR[1]);
  aZ[0] = wmma4(a, b[2], aZ[0]);
  aZ[1] = wmma4(a, b[3], aZ[1]);
  aN[0] = wmma4(a, b[4], aN[0]);
  aN[1] = wmma4(a, b[5], aN[1]);
}

// Three gate GEMMs (r, z, n) for this wave's two column tiles, with explicit
// 2-stage software pipelining: loads for stage k+1 are in flight while stage
// k's wmma chain executes. Extra stage costs 14 VGPRs; total ~145, no spills.
// wt: K-pair-interleaved weights: float2 element [p][n] = (W[n][2p], W[n][2p+1])
// ap: LDS activations:            float2 element [p][m] = (A[m][2p], A[m][2p+1])
__device__ __forceinline__ void gemm3x2(const v2f* __restrict__ wt,
                                        const v2f* __restrict__ ap, int ksteps,
                                        int nb, int col, int kh, v8f* aR,
                                        v8f* aZ, v8f* aN) {
  const v2f* a_p = ap + kh * 16 + col;                 // +2 pairs (K+4) per step
  const v2f* b_p = wt + (size_t)kh * G_DIM + nb + col; // +2 pair-rows per step
  v2f a0, a1, b0[6], b1[6];
  a0 = a_p[0];
  ldb6(b_p, b0);
  a_p += 32;
  b_p += 2 * G_DIM;
  int kk = 0;
#pragma unroll 1
  for (; kk + 2 <= ksteps - 1; kk += 2) {
    a1 = a_p[0];
    ldb6(b_p, b1);   // prefetch iter kk+1
    a_p += 32;
    b_p += 2 * G_DIM;
    mac6(a0, b0, aR, aZ, aN);  // compute iter kk
    a0 = a_p[0];
    ldb6(b_p, b0);   // prefetch iter kk+2
    a_p += 32;
    b_p += 2 * G_DIM;
    mac6(a1, b1, aR, aZ, aN);  // compute iter kk+1
  }
  if (kk + 1 < ksteps) {  // two iterations remain
    a1 = a_p[0];
    ldb6(b_p, b1);
    mac6(a0, b0, aR, aZ, aN);
    mac6(a1, b1, aR, aZ, aN);
  } else {  // one remains
    mac6(a0, b0, aR, aZ, aN);
  }
}

// GRU gating on this wave's two tiles: h = (1-z)*tanh(nx + r*nh) + z*h
__device__ __forceinline__ void gate_update(float* __restrict__ hF,
                                            const v8f* aR, const v8f* aZ,
                                            const v8f* aNx, const v8f* aNh,
                                            const float* br, const float* bz,
                                            const float* bnx, const float* bnh,
                                            int nb, int col, int kh) {
#pragma unroll
  for (int j = 0; j < 2; ++j) {
    const int n = nb + j * 16 + col;
    const int base = (n >> 1) * 32 + (n & 1);
#pragma unroll
    for (int e = 0; e < 8; ++e) {
      const int m = e + 8 * kh;  // C/D layout: vgpr e holds M=e (lanes 0-15), M=e+8 (16-31)
      float r = sigm(aR[j][e] + br[j]);
      float z = sigm(aZ[j][e] + bz[j]);
      float nn = tanh_fast(aNx[j][e] + bnx[j] + r * (aNh[j][e] + bnh[j]));
      const int idx = base + 2 * m;
      float hp = hF[idx];
      hF[idx] = (1.0f - z) * nn + z * hp;
    }
  }
}

__global__ void __launch_bounds__(THREADS, 1)
gru_decode_kernel(const float* __restrict__ x, const v2f* __restrict__ wt_ih0,
                  const v2f* __restrict__ wt_hh0, const float* __restrict__ b_ih0,
                  const float* __restrict__ b_hh0, const v2f* __restrict__ wt_ih1,
                  const v2f* __restrict__ wt_hh1, const float* __restrict__ b_ih1,
                  const float* __restrict__ b_hh1, const v2f* __restrict__ wt_out,
                  const float* __restrict__ b_out, float* __restrict__ out) {
  extern __shared__ v2f smem[];
  v2f* sx = smem;
  v2f* h0 = smem + SX_F2;
  v2f* h1 = h0 + H_F2;
  float* h0F = (float*)h0;
  float* h1F = (float*)h1;

  const int tid = threadIdx.x;
  const int w = tid >> 5;     // wave id, 0..15
  const int lane = tid & 31;
  const int col = lane & 15;  // N column / M row within tile
  const int kh = lane >> 4;   // half-wave K offset selector
  const int wg = blockIdx.x;
  const int nb = w * 32;      // this wave owns gate columns [nb, nb+32)

  // ---- load this block's x tile into the circular LDS window ----
  for (int i = tid; i < BM * T_SEQ * 94; i += THREADS) {
    int m = i & 15;
    int r = i >> 4;
    int p = r % 94;
    int t = r / 94;
    const float* xp = x + (((size_t)(wg * BM + m)) * T_SEQ + t) * Y_DIM + 2 * p;
    v2f v;
    v.x = xp[0];
    v.y = xp[1];
    sx[(t * 96 + p) * 16 + m] = v;
  }

  // ---- hoist biases into registers (per wave/lane/tile) ----
  float br0[2], bz0[2], bnx0[2], bnh0[2];
  float br1[2], bz1[2], bnx1[2], bnh1[2];
#pragma unroll
  for (int j = 0; j < 2; ++j) {
    int g = nb + j * 16 + col;
    br0[j] = b_ih0[g] + b_hh0[g];
    bz0[j] = b_ih0[H_DIM + g] + b_hh0[H_DIM + g];
    bnx0[j] = b_ih0[2 * H_DIM + g];
    bnh0[j] = b_hh0[2 * H_DIM + g];
    br1[j] = b_ih1[g] + b_hh1[g];
    bz1[j] = b_ih1[H_DIM + g] + b_hh1[H_DIM + g];
    bnx1[j] = b_ih1[2 * H_DIM + g];
    bnh1[j] = b_hh1[2 * H_DIM + g];
  }
  const int yo = w * 16 + col;  // output column for the projection phase
  const float bo = (w < 12 && yo < Y_DIM) ? b_out[yo] : 0.0f;

#pragma unroll 1
  for (int s = 0; s < NSTEP; ++s) {
    __syncthreads();  // previous step's output phase fully done (and x-load at s=0)
    for (int i = tid; i < 2 * H_F2; i += THREADS) {
      v2f zz;
      zz.x = 0.0f;
      zz.y = 0.0f;
      h0[i] = zz;  // h0,h1 contiguous: zero both
    }
    __syncthreads();

#pragma unroll 1
    for (int t = 0; t < T_SEQ; ++t) {
      const int slot = (s + t) % T_SEQ;
      // ---------- layer 0 ----------
      {
        v8f aR[2] = {}, aZ[2] = {}, aNx[2] = {}, aNh[2] = {};
        gemm3x2(wt_ih0, sx + slot * 96 * 16, 47, nb, col, kh, aR, aZ, aNx); // K=188
        gemm3x2(wt_hh0, h0, 128, nb, col, kh, aR, aZ, aNh);                 // K=512
        __syncthreads();
        gate_update(h0F, aR, aZ, aNx, aNh, br0, bz0, bnx0, bnh0, nb, col, kh);
        __syncthreads();
      }
      // ---------- layer 1 ----------
      {
        v8f cR[2] = {}, cZ[2] = {}, cNx[2] = {}, cNh[2] = {};
        gemm3x2(wt_ih1, h0, 128, nb, col, kh, cR, cZ, cNx);
        gemm3x2(wt_hh1, h1, 128, nb, col, kh, cR, cZ, cNh);
        __syncthreads();
        gate_update(h1F, cR, cZ, cNx, cNh, br1, bz1, bnx1, bnh1, nb, col, kh);
        __syncthreads();
      }
    }

    // ---------- output projection for last time step only ----------
    const int slotLast = (s + T_SEQ - 1) % T_SEQ;
    const int slotNew = s % T_SEQ;  // oldest slot gets overwritten by the new frame
    if (w < 12) {                   // 12 tiles cover Y padded to 192
      v8f acc = {};
      const v2f* a_p = h1 + kh * 16 + col;
      const v2f* b_p = wt_out + (size_t)kh * 192 + yo;
#pragma unroll 1
      for (int kk = 0; kk < 128; ++kk) {
        v2f a = *a_p;
        a.x = fmaxf(a.x, 0.0f);  // relu fused into A load
        a.y = fmaxf(a.y, 0.0f);
        acc = wmma4(a, *b_p, acc);
        a_p += 32;
        b_p += 2 * 192;
      }
      if (yo < Y_DIM) {
        const float* sxl = (const float*)(sx + slotLast * 96 * 16);
        float* sxn = (float*)(sx + slotNew * 96 * 16);
#pragma unroll
        for (int e = 0; e < 8; ++e) {
          const int m = e + 8 * kh;
          const int xi = (yo >> 1) * 32 + 2 * m + (yo & 1);
          float val = acc[e] + bo + sxl[xi];  // + residual x[:, -1, :]
          out[(((size_t)(wg * BM + m)) * NSTEP + s) * Y_DIM + yo] = val;
          sxn[xi] = val;  // append new frame to sliding window
        }
      }
    }
  }
}

// Pre-transpose weights into K-pair-interleaved [K/2][Nout]{k,k+1} layout so the
// WMMA B-operand global_load_b64 per half-wave touches exactly one 128B line.
__global__ void transpose_pair_kernel(const float* __restrict__ in,
                                      float* __restrict__ out, int Nin, int K,
                                      int Nout) {
  int idx = blockIdx.x * blockDim.x + threadIdx.x;
  int total = (K >> 1) * Nout;
  if (idx >= total) return;
  int n = idx % Nout;
  int p = idx / Nout;
  float a = 0.0f, b = 0.0f;
  if (n < Nin) {
    a = in[(size_t)n * K + 2 * p];
    b = in[(size_t)n * K + 2 * p + 1];
  }
  out[2 * (size_t)idx] = a;
  out[2 * (size_t)idx + 1] = b;
}

extern "C" void kernel_launch(void* const* d_in, const int* in_sizes, int n_in,
                              void* d_out, int out_size, void* d_ws,
                              size_t ws_size, hipStream_t stream) {
  (void)in_sizes; (void)n_in; (void)out_size; (void)ws_size;
  const float* x = (const float*)d_in[0];
  const float* W_ih0 = (const float*)d_in[1];
  const float* W_hh0 = (const float*)d_in[2];
  const float* b_ih0 = (const float*)d_in[3];
  const float* b_hh0 = (const float*)d_in[4];
  const float* W_ih1 = (const float*)d_in[5];
  const float* W_hh1 = (const float*)d_in[6];
  const float* b_ih1 = (const float*)d_in[7];
  const float* b_hh1 = (const float*)d_in[8];
  const float* W_out = (const float*)d_in[9];
  const float* b_out = (const float*)d_in[10];
  float* out = (float*)d_out;

  float* ws = (float*)d_ws;
  float* wt_ih0 = ws;
  size_t o = (size_t)94 * 1536 * 2;   // 188x1536
  float* wt_hh0 = ws + o;
  o += (size_t)256 * 1536 * 2;        // 512x1536
  float* wt_ih1 = ws + o;
  o += (size_t)256 * 1536 * 2;
  float* wt_hh1 = ws + o;
  o += (size_t)256 * 1536 * 2;
  float* wt_out = ws + o;             // 512x192 (Y padded, zero-filled)

  {
    int total = 94 * 1536;
    transpose_pair_kernel<<<(total + 255) / 256, 256, 0, stream>>>(
        W_ih0, wt_ih0, 1536, 188, 1536);
    total = 256 * 1536;
    transpose_pair_kernel<<<(total + 255) / 256, 256, 0, stream>>>(
        W_hh0, wt_hh0, 1536, 512, 1536);
    transpose_pair_kernel<<<(total + 255) / 256, 256, 0, stream>>>(
        W_ih1, wt_ih1, 1536, 512, 1536);
    transpose_pair_kernel<<<(total + 255) / 256, 256, 0, stream>>>(
        W_hh1, wt_hh1, 1536, 512, 1536);
    total = 256 * 192;
    transpose_pair_kernel<<<(total + 255) / 256, 256, 0, stream>>>(
        W_out, wt_out, 188, 512, 192);
  }

  size_t lds_bytes = (size_t)LDS_F2 * sizeof(v2f);  // 225,280 B
  gru_decode_kernel<<<B_TOT / BM, THREADS, lds_bytes, stream>>>(
      x, (const v2f*)wt_ih0, (const v2f*)wt_hh0, b_ih0, b_hh0,
      (const v2f*)wt_ih1, (const v2f*)wt_hh1, b_ih1, b_hh1, (const v2f*)wt_out,
      b_out, out);
}